// BackboneSolver_25941602468404
// MI455X (gfx1250) — compile-verified
//
#include <hip/hip_runtime.h>

typedef __attribute__((ext_vector_type(2))) float v2f;
typedef __attribute__((ext_vector_type(4))) float v4f;
typedef __attribute__((ext_vector_type(8))) float v8f;

namespace {
constexpr int kB = 16;
constexpr int kN = 4096;
constexpr int kK = 32;
constexpr int ROWS = 16;              // n-rows per wave
constexpr int WAVES = 8;              // waves per block
constexpr int G_PER_BLOCK = ROWS * WAVES;  // 128 rows/block
}

// Project 3x3 M onto SO(3):  R = U * diag(1,1,d) * V^T  (== reference rot_inv)
__device__ __forceinline__ void so3_project(const float m[9], float R[9]) {
  // S = M^T M (symmetric PSD)
  float S[3][3];
#pragma unroll
  for (int i = 0; i < 3; ++i)
#pragma unroll
    for (int j = 0; j < 3; ++j)
      S[i][j] = m[0 * 3 + i] * m[0 * 3 + j] + m[1 * 3 + i] * m[1 * 3 + j] +
                m[2 * 3 + i] * m[2 * 3 + j];
  float V[3][3] = {{1.f, 0.f, 0.f}, {0.f, 1.f, 0.f}, {0.f, 0.f, 1.f}};

  const int PP[3] = {0, 0, 1};
  const int QQ[3] = {1, 2, 2};
#pragma unroll
  for (int sweep = 0; sweep < 6; ++sweep) {
#pragma unroll
    for (int pq = 0; pq < 3; ++pq) {
      const int p = PP[pq];
      const int q = QQ[pq];
      float apq = S[p][q];
      if (fabsf(apq) > 1e-20f) {
        float tau = (S[q][q] - S[p][p]) / (2.0f * apq);
        float t = copysignf(1.0f, tau) / (fabsf(tau) + sqrtf(1.0f + tau * tau));
        float c = 1.0f / sqrtf(1.0f + t * t);
        float s = t * c;
#pragma unroll
        for (int k = 0; k < 3; ++k) {  // S = G^T S
          float sp = S[p][k], sq = S[q][k];
          S[p][k] = c * sp - s * sq;
          S[q][k] = s * sp + c * sq;
        }
#pragma unroll
        for (int k = 0; k < 3; ++k) {  // S = S G
          float sp = S[k][p], sq = S[k][q];
          S[k][p] = c * sp - s * sq;
          S[k][q] = s * sp + c * sq;
        }
#pragma unroll
        for (int k = 0; k < 3; ++k) {  // V = V G
          float vp = V[k][p], vq = V[k][q];
          V[k][p] = c * vp - s * vq;
          V[k][q] = s * vp + c * vq;
        }
      }
    }
  }

  float e0 = S[0][0], e1 = S[1][1], e2 = S[2][2];
  // sort eigenvalues descending, swapping V columns
  if (e0 < e1) {
    float t = e0; e0 = e1; e1 = t;
#pragma unroll
    for (int k = 0; k < 3; ++k) { float v = V[k][0]; V[k][0] = V[k][1]; V[k][1] = v; }
  }
  if (e0 < e2) {
    float t = e0; e0 = e2; e2 = t;
#pragma unroll
    for (int k = 0; k < 3; ++k) { float v = V[k][0]; V[k][0] = V[k][2]; V[k][2] = v; }
  }
  if (e1 < e2) {
    float t = e1; e1 = e2; e2 = t;
#pragma unroll
    for (int k = 0; k < 3; ++k) { float v = V[k][1]; V[k][1] = V[k][2]; V[k][2] = v; }
  }
  // force det(V) = +1 (negate last column if needed)
  float detV = V[0][0] * (V[1][1] * V[2][2] - V[2][1] * V[1][2]) -
               V[0][1] * (V[1][0] * V[2][2] - V[2][0] * V[1][2]) +
               V[0][2] * (V[1][0] * V[2][1] - V[2][0] * V[1][1]);
  if (detV < 0.0f) { V[0][2] = -V[0][2]; V[1][2] = -V[1][2]; V[2][2] = -V[2][2]; }

  // u0 = normalize(M v0); u1 = orthonormalized(M v1); u2 = u0 x u1
  float u0[3], u1[3], u2[3];
#pragma unroll
  for (int i = 0; i < 3; ++i)
    u0[i] = m[i * 3 + 0] * V[0][0] + m[i * 3 + 1] * V[1][0] + m[i * 3 + 2] * V[2][0];
#pragma unroll
  for (int i = 0; i < 3; ++i)
    u1[i] = m[i * 3 + 0] * V[0][1] + m[i * 3 + 1] * V[1][1] + m[i * 3 + 2] * V[2][1];
  float inv0 = 1.0f / sqrtf(fmaxf(u0[0] * u0[0] + u0[1] * u0[1] + u0[2] * u0[2], 1e-24f));
#pragma unroll
  for (int i = 0; i < 3; ++i) u0[i] *= inv0;
  float d01 = u0[0] * u1[0] + u0[1] * u1[1] + u0[2] * u1[2];
#pragma unroll
  for (int i = 0; i < 3; ++i) u1[i] -= d01 * u0[i];
  float inv1 = 1.0f / sqrtf(fmaxf(u1[0] * u1[0] + u1[1] * u1[1] + u1[2] * u1[2], 1e-24f));
#pragma unroll
  for (int i = 0; i < 3; ++i) u1[i] *= inv1;
  u2[0] = u0[1] * u1[2] - u0[2] * u1[1];
  u2[1] = u0[2] * u1[0] - u0[0] * u1[2];
  u2[2] = u0[0] * u1[1] - u0[1] * u1[0];

  // R = U V^T
#pragma unroll
  for (int i = 0; i < 3; ++i)
#pragma unroll
    for (int j = 0; j < 3; ++j)
      R[i * 3 + j] = u0[i] * V[j][0] + u1[i] * V[j][1] + u2[i] * V[j][2];
}

__global__ __launch_bounds__(WAVES * 32) void backbone_solver_kernel(
    const float* __restrict__ frames_rot, const float* __restrict__ frames_trans,
    const float* __restrict__ pair_rot, const float* __restrict__ pair_trans,
    const float* __restrict__ conf, const int* __restrict__ topo,
    float* __restrict__ out) {
  __shared__ float sh[WAVES][kK][16];    // per-wave 32(k) x 16(component) staging tile
  __shared__ float res[WAVES][ROWS][12]; // per-wave reduced results

  const int lane = threadIdx.x & 31;
  const int wave = threadIdx.x >> 5;
  const int base_g = blockIdx.x * G_PER_BLOCK + wave * ROWS;
  const int col = lane & 15;
  const int hi2 = (lane >> 4) << 1;  // 0 for lanes 0-15, 2 for lanes 16-31
  float* shw = &sh[wave][0][0];

  for (int r = 0; r < ROWS; ++r) {
    const int g = base_g + r;         // flat (b,n) index
    const int b = g >> 12;            // n = g & 4095
    const int off = g * kK + lane;    // flat (b,n,k) index, lane == k

    const int jn = topo[off];
    const float w = conf[off];
    const float* F = frames_rot + (size_t)((b << 12) + jn) * 9;
    const float* FT = frames_trans + (size_t)((b << 12) + jn) * 3;
    const float* P = pair_rot + (size_t)off * 9;
    const float* PT = pair_trans + (size_t)off * 3;

    const float f0 = F[0], f1 = F[1], f2 = F[2];
    const float f3 = F[3], f4 = F[4], f5 = F[5];
    const float f6 = F[6], f7 = F[7], f8 = F[8];
    const float p0 = P[0], p1 = P[1], p2 = P[2];
    const float p3 = P[3], p4 = P[4], p5 = P[5];
    const float p6 = P[6], p7 = P[7], p8 = P[8];
    const float t0 = PT[0], t1 = PT[1], t2 = PT[2];
    const float g0 = FT[0], g1 = FT[1], g2 = FT[2];

    float v[16];
    // comp_rot = F * P (row-major)
    v[0] = f0 * p0 + f1 * p3 + f2 * p6;
    v[1] = f0 * p1 + f1 * p4 + f2 * p7;
    v[2] = f0 * p2 + f1 * p5 + f2 * p8;
    v[3] = f3 * p0 + f4 * p3 + f5 * p6;
    v[4] = f3 * p1 + f4 * p4 + f5 * p7;
    v[5] = f3 * p2 + f4 * p5 + f5 * p8;
    v[6] = f6 * p0 + f7 * p3 + f8 * p6;
    v[7] = f6 * p1 + f7 * p4 + f8 * p7;
    v[8] = f6 * p2 + f7 * p5 + f8 * p8;
    // comp_trans = F * pt + ft
    v[9] = f0 * t0 + f1 * t1 + f2 * t2 + g0;
    v[10] = f3 * t0 + f4 * t1 + f5 * t2 + g1;
    v[11] = f6 * t0 + f7 * t1 + f8 * t2 + g2;
    v[12] = 0.f; v[13] = 0.f; v[14] = 0.f; v[15] = 0.f;

    // normalized confidence p = w / sum_k(w) via wave32 butterfly
    float ws = w;
    ws += __shfl_xor(ws, 16, 32);
    ws += __shfl_xor(ws, 8, 32);
    ws += __shfl_xor(ws, 4, 32);
    ws += __shfl_xor(ws, 2, 32);
    ws += __shfl_xor(ws, 1, 32);
    const float pw = w / ws;
#pragma unroll
    for (int c = 0; c < 12; ++c) v[c] *= pw;

    // stage premultiplied values: sh[k][c]
    v4f* dst = (v4f*)(shw + lane * 16);
    v4f d0 = {v[0], v[1], v[2], v[3]};
    v4f d1 = {v[4], v[5], v[6], v[7]};
    v4f d2 = {v[8], v[9], v[10], v[11]};
    v4f d3 = {0.f, 0.f, 0.f, 0.f};
    dst[0] = d0; dst[1] = d1; dst[2] = d2; dst[3] = d3;

    // Weighted sum over K=32 via 8 chained V_WMMA_F32_16X16X4_F32 with A == ones.
    // D[m][c] = sum_k 1 * B[k][c]; order of k within B slots is irrelevant.
    v2f a; a.x = 1.0f; a.y = 1.0f;
    v8f acc = {0.f, 0.f, 0.f, 0.f, 0.f, 0.f, 0.f, 0.f};
#pragma unroll
    for (int t = 0; t < 8; ++t) {
      v2f bv;
      bv.x = shw[(4 * t + hi2 + 0) * 16 + col];
      bv.y = shw[(4 * t + hi2 + 1) * 16 + col];
      acc = __builtin_amdgcn_wmma_f32_16x16x4_f32(
          /*neg_a=*/false, a, /*neg_b=*/false, bv,
          /*c_mod=*/(short)0, acc, /*reuse_a=*/false, /*reuse_b=*/false);
    }
    // D row 0 lives in acc element 0, N = lane (lanes 0..15)
    if (lane < 12) res[wave][r][lane] = acc[0];
  }

  // Phase 2: lane L handles row L's SVD projection (same-wave LDS is in-order)
  if (lane < ROWS) {
    const v4f* rsrc = (const v4f*)&res[wave][lane][0];
    v4f q0 = rsrc[0], q1 = rsrc[1], q2 = rsrc[2];
    float m9[9] = {q0.x, q0.y, q0.z, q0.w, q1.x, q1.y, q1.z, q1.w, q2.x};
    float R[9];
    so3_project(m9, R);

    const int g = base_g + lane;
    v4f* o = (v4f*)(out + (size_t)g * 12);
    v4f o0 = {R[0], R[1], R[2], R[3]};
    v4f o1 = {R[4], R[5], R[6], R[7]};
    v4f o2 = {R[8], q2.y, q2.z, q2.w};  // rot_inv(9) + avg_trans(3)
    o[0] = o0; o[1] = o1; o[2] = o2;
  }
}

extern "C" void kernel_launch(void* const* d_in, const int* in_sizes, int n_in,
                              void* d_out, int out_size, void* d_ws, size_t ws_size,
                              hipStream_t stream) {
  const float* frames_rot = (const float*)d_in[0];
  const float* frames_trans = (const float*)d_in[1];
  const float* pair_rot = (const float*)d_in[2];
  const float* pair_trans = (const float*)d_in[3];
  const float* conf = (const float*)d_in[4];
  const int* topo = (const int*)d_in[5];
  float* out = (float*)d_out;

  const int total = kB * kN;                 // 65536 rows
  dim3 grid(total / G_PER_BLOCK);            // 512 blocks
  dim3 block(WAVES * 32);                    // 256 threads = 8 waves
  hipLaunchKernelGGL(backbone_solver_kernel, grid, block, 0, stream,
                     frames_rot, frames_trans, pair_rot, pair_trans, conf, topo,
                     out);
}